// OmegaSSMLayer_13305808683218
// MI455X (gfx1250) — compile-verified
//
#include <hip/hip_runtime.h>
#include <hip/hip_bf16.h>

#define D_MODEL   1024
#define SEQ_L     4096
#define BATCH     16
#define OMEGA_DIM 64
#define N_PARAMS  2016
#define VQ_CODES  256
#define VQ_DIM    64
#define LSPLIT    8

typedef __attribute__((ext_vector_type(2))) float v2f;
typedef __attribute__((ext_vector_type(8))) float v8f;

__device__ __forceinline__ v8f wmma_f32_4(v2f a, v2f b, v8f c) {
    // D = A(16x4 f32) * B(4x16 f32) + C(16x16 f32)
    return __builtin_amdgcn_wmma_f32_16x16x4_f32(
        /*neg_a=*/false, a, /*neg_b=*/false, b,
        /*c_mod=*/(short)0, c, /*reuse_a=*/false, /*reuse_b=*/false);
}

// ---------------------------------------------------------------------------
// K1: partial column sums of x over L (LSPLIT chunks). part[z][b][d].
// ---------------------------------------------------------------------------
__launch_bounds__(256)
__global__ void omega_k1_colsum(const float* __restrict__ x, float* __restrict__ part) {
    const int b = blockIdx.x;
    const int d = blockIdx.y * 256 + threadIdx.x;
    const int z = blockIdx.z;
    const int lchunk = SEQ_L / LSPLIT; // 512
    const float* base = x + ((size_t)b * SEQ_L + (size_t)z * lchunk) * D_MODEL + d;
    float s = 0.f;
    #pragma unroll 8
    for (int l = 0; l < lchunk; ++l) s += base[(size_t)l * D_MODEL];
    part[((size_t)z * BATCH + b) * D_MODEL + d] = s;
}

// ---------------------------------------------------------------------------
// K2: per-batch small math. One block (256 thr) per batch.
//   h_mean -> params -> Omega -> G = I + O + O^2/2 + O^3/6 -> ws_G
//   h_mean -> vq_input -> argmax code -> vq_contribution -> ws_vqc
// ---------------------------------------------------------------------------
__launch_bounds__(256)
__global__ void omega_k2_batch(const float* __restrict__ part,
                               const float* __restrict__ opw, const float* __restrict__ opb,
                               const float* __restrict__ viw, const float* __restrict__ vib,
                               const float* __restrict__ vow, const float* __restrict__ vob,
                               const float* __restrict__ cb,
                               float* __restrict__ Gws, float* __restrict__ vqcws) {
    __shared__ float hm[D_MODEL];
    __shared__ float params[N_PARAMS];
    __shared__ float om [OMEGA_DIM * OMEGA_DIM];
    __shared__ float om2[OMEGA_DIM * OMEGA_DIM];
    __shared__ float om3[OMEGA_DIM * OMEGA_DIM];
    __shared__ float vqin[VQ_DIM];
    __shared__ float scores[VQ_CODES];
    __shared__ int   sidx;

    const int b = blockIdx.x, tid = threadIdx.x;

    // finish mean
    for (int d = tid; d < D_MODEL; d += 256) {
        float s = 0.f;
        #pragma unroll
        for (int z = 0; z < LSPLIT; ++z) s += part[((size_t)z * BATCH + b) * D_MODEL + d];
        hm[d] = s * (1.f / (float)SEQ_L);
    }
    __syncthreads();

    // params = h_mean @ opw.T + opb
    for (int p = tid; p < N_PARAMS; p += 256) {
        const float* wr = opw + (size_t)p * D_MODEL;
        float s = opb[p];
        for (int d = 0; d < D_MODEL; ++d) s += hm[d] * wr[d];
        params[p] = s;
    }
    for (int i = tid; i < OMEGA_DIM * OMEGA_DIM; i += 256) om[i] = 0.f;
    __syncthreads();

    // scatter triu params into skew-symmetric omega
    {
        int p = 0;
        for (int r = 0; r < OMEGA_DIM - 1; ++r)
            for (int c = r + 1; c < OMEGA_DIM; ++c) {
                if ((p & 255) == tid) {
                    float v = params[p];
                    om[r * OMEGA_DIM + c] = v;
                    om[c * OMEGA_DIM + r] = -v;
                }
                ++p;
            }
    }
    __syncthreads();

    // om2 = om @ om
    for (int idx = tid; idx < OMEGA_DIM * OMEGA_DIM; idx += 256) {
        int i = idx >> 6, j = idx & 63;
        float s = 0.f;
        for (int k = 0; k < OMEGA_DIM; ++k) s += om[i * 64 + k] * om[k * 64 + j];
        om2[idx] = s;
    }
    __syncthreads();
    // om3 = om2 @ om
    for (int idx = tid; idx < OMEGA_DIM * OMEGA_DIM; idx += 256) {
        int i = idx >> 6, j = idx & 63;
        float s = 0.f;
        for (int k = 0; k < OMEGA_DIM; ++k) s += om2[i * 64 + k] * om[k * 64 + j];
        om3[idx] = s;
    }
    __syncthreads();
    // G = I + om + om2/2 + om3/6
    for (int idx = tid; idx < OMEGA_DIM * OMEGA_DIM; idx += 256) {
        int i = idx >> 6, j = idx & 63;
        float g = ((i == j) ? 1.f : 0.f) + om[idx] + 0.5f * om2[idx] + (1.f / 6.f) * om3[idx];
        Gws[(size_t)b * OMEGA_DIM * OMEGA_DIM + idx] = g;
    }

    // ---- VQ path ----
    if (tid < VQ_DIM) {
        const float* wr = viw + (size_t)tid * D_MODEL;
        float s = vib[tid];
        for (int d = 0; d < D_MODEL; ++d) s += hm[d] * wr[d];
        vqin[tid] = s;
    }
    __syncthreads();
    {
        // argmax over cosine scores; vq_input norm is a positive constant -> drop it
        const float* c = cb + (size_t)tid * VQ_DIM;
        float dot = 0.f, nn = 0.f;
        for (int e = 0; e < VQ_DIM; ++e) { dot += vqin[e] * c[e]; nn += c[e] * c[e]; }
        scores[tid] = dot / fmaxf(sqrtf(nn), 1e-12f);
    }
    __syncthreads();
    if (tid == 0) {
        float best = scores[0]; int bi = 0;
        for (int c = 1; c < VQ_CODES; ++c)
            if (scores[c] > best) { best = scores[c]; bi = c; }
        sidx = bi;
    }
    __syncthreads();
    // vq_out == codebook[idx] (straight-through); contribution = code @ vow.T + vob
    {
        const float* code = cb + (size_t)sidx * VQ_DIM;
        for (int d = tid; d < D_MODEL; d += 256) {
            const float* wr = vow + (size_t)d * VQ_DIM;
            float s = vob[d];
            #pragma unroll
            for (int e = 0; e < VQ_DIM; ++e) s += code[e] * wr[e];
            vqcws[(size_t)b * D_MODEL + d] = s;
        }
    }
}

// ---------------------------------------------------------------------------
// K3: fused main pass. One block per 16-row L-tile (256 thr = 8 wave32).
//   xrot = x_head @ G           (wave 0, 64x v_wmma_f32_16x16x4_f32)
//   lie  = xrot @ omega_out_w.T (8 waves x 8 n-tiles x 16 k-steps WMMA)
//   pre  = x + om*(lie+b) + vm*vqc  -> LDS slab -> rmsnorm -> out
// ---------------------------------------------------------------------------
__launch_bounds__(256)
__global__ void omega_k3_main(const float* __restrict__ x,
                              const float* __restrict__ wout,   // (1024,64)
                              const float* __restrict__ ob,     // (1024,)
                              const float* __restrict__ Gws,
                              const float* __restrict__ vqcws,
                              const float* __restrict__ nw,
                              const float* __restrict__ p_om, const float* __restrict__ p_vm,
                              float* __restrict__ out) {
    extern __shared__ float lds[];
    float* Gs   = lds;                    // 4096
    float* xrot = Gs + 4096;              // 16*64 = 1024
    float* vqcs = xrot + 1024;            // 1024
    float* slab = vqcs + 1024;            // 16*1024 = 16384
    float* red  = slab + 16384;           // 256
    float* rsr  = red + 256;              // 16

    const int b    = blockIdx.x >> 8;     // 256 tiles per batch
    const int tile = blockIdx.x & 255;
    const int l0   = tile * 16;
    const int tid  = threadIdx.x;
    const int lane = tid & 31;
    const int wave = tid >> 5;
    const int nIdx = lane & 15;           // N / M index within fragment
    const int koff = (lane >> 4) << 1;    // 0 or 2: upper lane half holds K=2,3
    const int mbase = (lane >> 4) * 8;    // C/D: rows 0-7 vs 8-15

    for (int i = tid; i < 4096; i += 256) Gs[i] = Gws[(size_t)b * 4096 + i];
    for (int i = tid; i < 1024; i += 256) vqcs[i] = vqcws[(size_t)b * 1024 + i];
    __syncthreads();

    const float om = p_om[0], vm = p_vm[0];
    const float* xrow = x + ((size_t)b * SEQ_L + l0) * D_MODEL;

    // --- GEMM1: xrot(16x64) = x_head(16x64) @ G(64x64), wave 0 only ---
    if (wave == 0) {
        const int m = nIdx;
        #pragma unroll
        for (int nt = 0; nt < 4; ++nt) {
            const int n0 = nt * 16;
            v8f acc = {};
            #pragma unroll
            for (int ks = 0; ks < 16; ++ks) {
                const int k = ks * 4 + koff;
                v2f a; a.x = xrow[(size_t)m * D_MODEL + k];
                       a.y = xrow[(size_t)m * D_MODEL + k + 1];
                v2f bb; bb.x = Gs[k * 64 + n0 + nIdx];
                        bb.y = Gs[(k + 1) * 64 + n0 + nIdx];
                acc = wmma_f32_4(a, bb, acc);
            }
            #pragma unroll
            for (int r = 0; r < 8; ++r)
                xrot[(mbase + r) * 64 + n0 + nIdx] = acc[r];
        }
    }
    __syncthreads();

    // preload A fragments (xrot) for GEMM2: 16 k-steps x 2 dwords/lane
    v2f a2[16];
    {
        const int m = nIdx;
        #pragma unroll
        for (int ks = 0; ks < 16; ++ks) {
            const int k = ks * 4 + koff;
            a2[ks].x = xrot[m * 64 + k];
            a2[ks].y = xrot[m * 64 + k + 1];
        }
    }

    // --- GEMM2: lie(16x1024) = xrot(16x64) @ wout.T(64x1024), 8 n-tiles/wave ---
    #pragma unroll 1
    for (int t = 0; t < 8; ++t) {
        const int d0 = (wave * 8 + t) * 16;
        const int d  = d0 + nIdx;
        v8f acc = {};
        #pragma unroll
        for (int ks = 0; ks < 16; ++ks) {
            const int k = ks * 4 + koff;
            v2f bb; bb.x = wout[(size_t)d * 64 + k];      // B[k][n] = wout[n][k]
                    bb.y = wout[(size_t)d * 64 + k + 1];
            acc = wmma_f32_4(a2[ks], bb, acc);
        }
        const float obd = ob[d];
        const float vqd = vqcs[d];
        #pragma unroll
        for (int r = 0; r < 8; ++r) {
            const int row = mbase + r;
            float pre = xrow[(size_t)row * D_MODEL + d] + om * (acc[r] + obd) + vm * vqd;
            slab[row * 1024 + d] = pre;
        }
    }
    __syncthreads();

    // --- rmsnorm over each of the 16 rows ---
    {
        const int row = tid >> 4, part = tid & 15;
        float s = 0.f;
        #pragma unroll 4
        for (int j = 0; j < 64; ++j) {
            float v = slab[row * 1024 + part + j * 16];
            s += v * v;
        }
        red[tid] = s;
    }
    __syncthreads();
    if (tid < 16) {
        float s = 0.f;
        #pragma unroll
        for (int j = 0; j < 16; ++j) s += red[tid * 16 + j];
        rsr[tid] = rsqrtf(s * (1.f / (float)D_MODEL) + 1e-6f);
    }
    __syncthreads();
    float* outbase = out + ((size_t)b * SEQ_L + l0) * D_MODEL;
    for (int idx = tid; idx < 16 * D_MODEL; idx += 256) {
        const int row = idx >> 10, d = idx & 1023;
        outbase[(size_t)row * D_MODEL + d] = slab[idx] * rsr[row] * nw[d];
    }
}

// ---------------------------------------------------------------------------
extern "C" void kernel_launch(void* const* d_in, const int* in_sizes, int n_in,
                              void* d_out, int out_size, void* d_ws, size_t ws_size,
                              hipStream_t stream) {
    const float* x    = (const float*)d_in[0];
    const float* opw  = (const float*)d_in[1];
    const float* opb  = (const float*)d_in[2];
    const float* oow  = (const float*)d_in[3];
    const float* oob  = (const float*)d_in[4];
    const float* viw  = (const float*)d_in[5];
    const float* vib  = (const float*)d_in[6];
    const float* vow  = (const float*)d_in[7];
    const float* vob  = (const float*)d_in[8];
    const float* cb   = (const float*)d_in[9];
    const float* omix = (const float*)d_in[10];
    const float* vmix = (const float*)d_in[11];
    const float* nw   = (const float*)d_in[12];
    float* out = (float*)d_out;

    float* wsf  = (float*)d_ws;
    float* part = wsf;                                   // LSPLIT*B*D = 131072 f
    float* Gws  = wsf + (size_t)LSPLIT * BATCH * D_MODEL; // B*64*64 = 65536 f
    float* vqc  = Gws + (size_t)BATCH * OMEGA_DIM * OMEGA_DIM; // B*D = 16384 f

    dim3 g1(BATCH, D_MODEL / 256, LSPLIT);
    omega_k1_colsum<<<g1, 256, 0, stream>>>(x, part);
    omega_k2_batch<<<BATCH, 256, 0, stream>>>(part, opw, opb, viw, vib, vow, vob,
                                              cb, Gws, vqc);
    const size_t shmem = (size_t)(4096 + 1024 + 1024 + 16384 + 256 + 16) * sizeof(float);
    omega_k3_main<<<BATCH * (SEQ_L / 16), 256, shmem, stream>>>(
        x, oow, oob, Gws, vqc, nw, omix, vmix, out);
}